// SimpleAverageFusion_30915174596612
// MI455X (gfx1250) — compile-verified
//
#include <hip/hip_runtime.h>
#include <hip/hip_bf16.h>

// ---------------------------------------------------------------------------
// MI455X (gfx1250) multi-view feature sampling.
//
//  * Projection (points @ (R*K) + T*K, /z) -> two V_WMMA_F32_16X16X4_F32:
//    A = 16 points x [x,y,z,1]; B = fused 4x3 matrices packed 4x16
//    (8 views x {x,y} columns, second WMMA for the 8 z columns).
//  * Coords are wave-uniform per (point,view): v_readlane -> SGPRs, scalar
//    early-out branch for the ~75% of view-samples outside [-1,1] (they
//    contribute exactly 0 under zeros-padding).
//  * Gather roofline: native [C,H,W] layout wastes ~94% of every 128B line
//    (channels 64KB apart). If ws_size allows, a tiled-LDS transpose to
//    channels-last [V,B,H,W,C] (256MB streaming ~ 11us at 23.3TB/s) makes each
//    corner 256B contiguous -> 1 fully-used 128B line per wave per corner,
//    cutting gather-side traffic from ~4-6GB of sparse line requests to
//    ~0.4GB coalesced. Fallback: direct gather (L2 still captures reuse,
//    32 maps x 4MB = 128MB < 192MB L2).
// ---------------------------------------------------------------------------

typedef float v2f __attribute__((ext_vector_type(2)));
typedef float v8f __attribute__((ext_vector_type(8)));

namespace svf {
constexpr int B_ = 4, N_ = 50000, V_ = 8, C_ = 64, H_ = 128, W_ = 128;
constexpr int HW_ = H_ * W_;
constexpr size_t FM_ELEMS = (size_t)V_ * B_ * C_ * HW_;
constexpr int PTS_PER_WAVE = 16;                   // one WMMA tile of points
constexpr int NGROUPS = (B_ * N_) / PTS_PER_WAVE;  // 12500 (divides exactly)
constexpr int CSPLIT = 2;                          // 2 waves cover 64 channels
constexpr int NWAVES = NGROUPS * CSPLIT;           // 25000
constexpr int WAVES_PER_BLOCK = 8;                 // 256 threads
constexpr int NBLOCKS = NWAVES / WAVES_PER_BLOCK;  // 3125 exactly
constexpr int TP_ = 32;                            // transpose tile: 32 px x 64 ch
constexpr int TBLOCKS = (V_ * B_) * (HW_ / TP_);   // 32 * 512 = 16384
}  // namespace svf

using namespace svf;

__device__ __forceinline__ float lane_bcast(float v, int lane) {
  // wave-uniform broadcast -> SGPR (enables scalar branches / SADDR bases)
#if __has_builtin(__builtin_amdgcn_readlane)
  return __builtin_bit_cast(
      float, __builtin_amdgcn_readlane(__builtin_bit_cast(int, v), lane));
#else
  return __shfl(v, lane, 32);
#endif
}

__device__ __forceinline__ float fused_m4(const float* __restrict__ Rm,
                                          const float* __restrict__ Tm,
                                          const float* __restrict__ Km,
                                          int b, int v, int k, int j) {
  // M4[v][b]: 4x3 matrix s.t. [x,y,z,1] @ M4 = points_img (pre-divide):
  // rows 0..2 = R*K, row 3 = T*K.
  const size_t vb = (size_t)v * B_ + b;
  const float* Kp = Km + vb * 9;
  const float kc0 = Kp[0 + j], kc1 = Kp[3 + j], kc2 = Kp[6 + j];
  float r0, r1, r2;
  if (k < 3) {
    const float* Rp = Rm + vb * 9 + (size_t)k * 3;
    r0 = Rp[0]; r1 = Rp[1]; r2 = Rp[2];
  } else {
    const float* Tp = Tm + vb * 3;
    r0 = Tp[0]; r1 = Tp[1]; r2 = Tp[2];
  }
  return fmaf(r0, kc0, fmaf(r1, kc1, r2 * kc2));
}

// fm [V,B,C,H,W] -> fmT [V,B,H,W,C]; tiled through LDS, both sides coalesced.
__global__ __launch_bounds__(256) void transpose_chlast_kernel(
    const float* __restrict__ in, float* __restrict__ out) {
  __shared__ float tile[C_][TP_ + 1];  // +1 pad: stride 33 is odd -> no conflicts
  const int vb = blockIdx.x >> 9;          // HW/TP = 512 tiles per (v,b)
  const int p0 = (blockIdx.x & 511) * TP_;
  const int t = threadIdx.x;
  const int pr = t & 31;                   // pixel within tile (contiguous reads)
  const int cr = t >> 5;                   // 0..7
  const float* src = in + (size_t)vb * C_ * HW_ + p0 + pr;
#pragma unroll
  for (int i = 0; i < 8; ++i) {
    const int cc = cr + 8 * i;
    // read-once stream: non-temporal so it doesn't evict the growing fmT in L2
    tile[cc][pr] = __builtin_nontemporal_load(src + (size_t)cc * HW_);
  }
  __syncthreads();
  float* dst = out + ((size_t)vb * HW_ + p0) * C_;
#pragma unroll
  for (int i = 0; i < 8; ++i) {
    const int idx = t + 256 * i;
    const int cw = idx & 63;                // contiguous in memory
    const int pw = idx >> 6;
    dst[(size_t)pw * C_ + cw] = tile[cw][pw];
  }
}

template <bool CHLAST>
__global__ __launch_bounds__(256) void project_sample_kernel(
    const float* __restrict__ pts,   // [B,N,3]
    const float* __restrict__ fms,   // CHLAST ? [V,B,H,W,C] : [V,B,C,H,W]
    const float* __restrict__ Rm,    // [V,B,3,3]
    const float* __restrict__ Tm,    // [V,B,3]
    const float* __restrict__ Km,    // [V,B,3,3]
    float* __restrict__ out_agg)     // [B,N,C]
{
  const int lane = threadIdx.x & 31;
  const int wid = blockIdx.x * WAVES_PER_BLOCK + (threadIdx.x >> 5);
  if (wid >= NWAVES) return;  // wave-uniform: EXEC stays all-ones for WMMA

  const int group = wid >> 1;  // 16-point group
  const int csel = wid & 1;    // which half of the channels
  const int pbase = group * PTS_PER_WAVE;
  const int b = pbase / N_;    // groups never straddle a batch (N%16==0)
  const int n0 = pbase % N_;

  // ---- A: 16 points x [x,y,z,1] (16x4 f32). ISA layout: lanes 0-15 hold
  //      K=0 (v0)/K=1 (v1); lanes 16-31 hold K=2/K=3.
  const int mrow = lane & 15;
  const float* P = pts + ((size_t)b * N_ + (n0 + mrow)) * 3;
  v2f A;
  if (lane < 16) { A.x = P[0]; A.y = P[1]; }
  else           { A.x = P[2]; A.y = 1.0f; }

  // ---- B: 4x16 f32 mirroring the A layout (lanes 0-15 rows K=0/1,
  //      lanes 16-31 rows K=2/3). B_xy col n: view n>>1, comp n&1.
  //      B_z col n (n<8): z row of view n; cols 8..15 zero.
  const int ncol = lane & 15;
  const int vxy = ncol >> 1, comp = ncol & 1;
  const int k0 = (lane < 16) ? 0 : 2;
  v2f Bxy, Bz;
  Bxy.x = fused_m4(Rm, Tm, Km, b, vxy, k0, comp);
  Bxy.y = fused_m4(Rm, Tm, Km, b, vxy, k0 + 1, comp);
  if (ncol < 8) {
    Bz.x = fused_m4(Rm, Tm, Km, b, ncol, k0, 2);
    Bz.y = fused_m4(Rm, Tm, Km, b, ncol, k0 + 1, 2);
  } else {
    Bz.x = 0.0f; Bz.y = 0.0f;
  }

  // ---- D = A x B ----
  v8f zc = {};
  v8f Dxy = __builtin_amdgcn_wmma_f32_16x16x4_f32(
      false, A, false, Bxy, (short)0, zc, false, false);
  v8f Dz = __builtin_amdgcn_wmma_f32_16x16x4_f32(
      false, A, false, Bz, (short)0, zc, false, false);

  // D layout: VGPR r, lane L -> point M = r + 8*(L/16), column N = L%16.
  // z for (M, view vxy) sits at column vxy of the same lane-half (per-lane
  // index -> ds_bpermute). Convert straight to pixel space (W==H==128).
  v8f Cpix;
#pragma unroll
  for (int r = 0; r < 8; ++r) {
    float zv = __shfl(Dz[r], (ncol >> 1) + (lane & 16), 32);
    float cn = Dxy[r] / zv;
    Cpix[r] = fmaf(cn + 1.0f, 128.0f, -1.0f) * 0.5f;
  }

  // ---- Bilinear gather: lane = channel, 16 points x 8 views ----
  const int c = csel * 32 + lane;
#pragma unroll
  for (int h = 0; h < 2; ++h) {
#pragma unroll
    for (int r = 0; r < 8; ++r) {
      const int m = r + 8 * h;
      float acc = 0.0f;
      for (int v = 0; v < V_; ++v) {
        // wave-uniform coords -> SGPRs (scalar skip branch, SADDR bases)
        const float x = lane_bcast(Cpix[r], 2 * v + 16 * h);
        const float y = lane_bcast(Cpix[r], 2 * v + 1 + 16 * h);
        const float x0f = floorf(x), y0f = floorf(y);
        const int ix0 = (int)x0f, iy0 = (int)y0f;
        if (ix0 < -1 || ix0 > W_ - 1 || iy0 < -1 || iy0 > H_ - 1) continue;
        const float wx = x - x0f, wy = y - y0f;
        const float mx0 = (ix0 >= 0) ? 1.0f : 0.0f;
        const float mx1 = (ix0 < W_ - 1) ? 1.0f : 0.0f;
        const float my0 = (iy0 >= 0) ? 1.0f : 0.0f;
        const float my1 = (iy0 < H_ - 1) ? 1.0f : 0.0f;
        const int ix0c = ix0 < 0 ? 0 : ix0;
        const int ix1c = ix0 + 1 > W_ - 1 ? W_ - 1 : ix0 + 1;
        const int iy0c = iy0 < 0 ? 0 : iy0;
        const int iy1c = iy0 + 1 > H_ - 1 ? H_ - 1 : iy0 + 1;
        const size_t vb = (size_t)v * B_ + b;
        float f00, f10, f01, f11;
        if (CHLAST) {
          // corner = 256B contiguous: each wave reads one full 128B line of it
          const float* base = fms + (size_t)vb * HW_ * C_;
          f00 = base[((size_t)(iy0c * W_ + ix0c)) * C_ + c];
          f10 = base[((size_t)(iy0c * W_ + ix1c)) * C_ + c];
          f01 = base[((size_t)(iy1c * W_ + ix0c)) * C_ + c];
          f11 = base[((size_t)(iy1c * W_ + ix1c)) * C_ + c];
        } else {
          const float* pc = fms + ((size_t)vb * C_ + c) * HW_;
          const float* row0 = pc + iy0c * W_;
          const float* row1 = pc + iy1c * W_;
          f00 = row0[ix0c]; f10 = row0[ix1c];
          f01 = row1[ix0c]; f11 = row1[ix1c];
        }
        const float w00 = (1.0f - wx) * (1.0f - wy) * mx0 * my0;
        const float w10 = wx * (1.0f - wy) * mx1 * my0;
        const float w01 = (1.0f - wx) * wy * mx0 * my1;
        const float w11 = wx * wy * mx1 * my1;
        acc = fmaf(f00, w00, acc);
        acc = fmaf(f10, w10, acc);
        acc = fmaf(f01, w01, acc);
        acc = fmaf(f11, w11, acc);
      }
      // mean over 8 views; 128B coalesced, never re-read -> non-temporal
      __builtin_nontemporal_store(
          acc * 0.125f, &out_agg[((size_t)b * N_ + (n0 + m)) * C_ + c]);
    }
  }
}

__global__ void fill_attn_kernel(float* __restrict__ p, int n, float val) {
  int i = blockIdx.x * blockDim.x + threadIdx.x;
  int stride = gridDim.x * blockDim.x;
  for (; i < n; i += stride) __builtin_nontemporal_store(val, &p[i]);
}

extern "C" void kernel_launch(void* const* d_in, const int* in_sizes, int n_in,
                              void* d_out, int out_size, void* d_ws, size_t ws_size,
                              hipStream_t stream) {
  (void)in_sizes; (void)n_in; (void)out_size;
  const float* pts = (const float*)d_in[0];  // [B,N,3]
  const float* fm  = (const float*)d_in[1];  // [V,B,C,H,W]
  const float* Rm  = (const float*)d_in[2];  // [V,B,3,3]
  const float* Tm  = (const float*)d_in[3];  // [V,B,3]
  const float* Km  = (const float*)d_in[4];  // [V,B,3,3]
  float* out_agg  = (float*)d_out;                         // [B,N,C]
  float* out_attn = (float*)d_out + (size_t)B_ * N_ * C_;  // [B,N,V]

  const size_t fm_bytes = FM_ELEMS * sizeof(float);  // 128 MB
  if (ws_size >= fm_bytes) {
    float* fmT = (float*)d_ws;
    transpose_chlast_kernel<<<TBLOCKS, 256, 0, stream>>>(fm, fmT);
    project_sample_kernel<true><<<NBLOCKS, WAVES_PER_BLOCK * 32, 0, stream>>>(
        pts, fmT, Rm, Tm, Km, out_agg);
  } else {
    project_sample_kernel<false><<<NBLOCKS, WAVES_PER_BLOCK * 32, 0, stream>>>(
        pts, fm, Rm, Tm, Km, out_agg);
  }

  const int attn_n = B_ * N_ * V_;
  fill_attn_kernel<<<(attn_n + 255) / 256, 256, 0, stream>>>(
      out_attn, attn_n, 1.0f / (float)V_);
}